// GeometricAwareFeatureAggregation_27204322853231
// MI455X (gfx1250) — compile-verified
//
#include <hip/hip_runtime.h>
#include <hip/hip_bf16.h>

#define N_PTS 65536
#define M     256
#define KNN1  10
#define KNN2  90
#define FDIM  768
#define NLBL  64

typedef float v2f __attribute__((ext_vector_type(2)));
typedef float v8f __attribute__((ext_vector_type(8)));
typedef unsigned int v4u __attribute__((ext_vector_type(4)));
typedef int v4i __attribute__((ext_vector_type(4)));
typedef int v8i __attribute__((ext_vector_type(8)));

// ---------------------------------------------------------------------------
// K0: zero the per-center counters (re-run every launch; graph-replay safe)
// ---------------------------------------------------------------------------
__global__ void zero_count_kernel(int* __restrict__ count) {
    count[threadIdx.x] = 0;
}

// ---------------------------------------------------------------------------
// K1: farthest point sampling. Single persistent workgroup (1024 threads);
// the 65536-entry running min-distance array lives entirely in LDS (256 KB of
// the WGP's 320 KB) so the 256-step serial chain never touches global memory
// except for the xyz reads. Interleaved indexing -> conflict-free LDS banks;
// the (val,idx) tree reduce breaks ties on smallest index (jnp.argmax).
// ---------------------------------------------------------------------------
__global__ __launch_bounds__(1024)
void fps_kernel(const float* __restrict__ pc, int* __restrict__ idx,
                float* __restrict__ sp) {
    __shared__ float s_d[N_PTS];          // 256 KB running min distances
    __shared__ float s_v[1024];
    __shared__ int   s_i[1024];
    __shared__ float s_px, s_py, s_pz;
    __shared__ int   s_last;
    const int tid = threadIdx.x;
    for (int p = tid; p < N_PTS; p += 1024) s_d[p] = __builtin_inff();
    if (tid == 0) s_last = 0;
    __syncthreads();

    for (int s = 0; s < M; ++s) {
        const int last = s_last;
        if (tid == 0) {
            idx[s] = last;
            float x = pc[last * 6 + 0], y = pc[last * 6 + 1], z = pc[last * 6 + 2];
            sp[s * 3 + 0] = x; sp[s * 3 + 1] = y; sp[s * 3 + 2] = z;
            s_px = x; s_py = y; s_pz = z;
        }
        __syncthreads();
        const float px = s_px, py = s_py, pz = s_pz;
        float best = -1.0f; int bidx = tid;
        for (int p = tid; p < N_PTS; p += 1024) {
            const float dx = pc[p * 6 + 0] - px;
            const float dy = pc[p * 6 + 1] - py;
            const float dz = pc[p * 6 + 2] - pz;
            const float d  = dx * dx + dy * dy + dz * dz;
            const float nd = fminf(s_d[p], d);
            s_d[p] = nd;
            if (nd > best) { best = nd; bidx = p; }   // strict > keeps first max
        }
        s_v[tid] = best; s_i[tid] = bidx;
        __syncthreads();
        for (int st = 512; st > 0; st >>= 1) {
            if (tid < st) {
                const float v2 = s_v[tid + st]; const int i2 = s_i[tid + st];
                if (v2 > s_v[tid] || (v2 == s_v[tid] && i2 < s_i[tid])) {
                    s_v[tid] = v2; s_i[tid] = i2;
                }
            }
            __syncthreads();
        }
        if (tid == 0) s_last = s_i[0];
        __syncthreads();
    }
}

// ---------------------------------------------------------------------------
// K2: up[i] = argmin_c ||xyz_i - sp_c||^2 (first-index tie-break) plus
// count[c] histogram. sp (3 KB) is staged to LDS with the CDNA5 async
// global->LDS path when the toolchain exposes it.
// ---------------------------------------------------------------------------
__global__ __launch_bounds__(256)
void up_count_kernel(const float* __restrict__ pc, const float* __restrict__ sp,
                     int* __restrict__ up, int* __restrict__ count) {
    __shared__ float s_sp[M * 3];
    const int tid = threadIdx.x;
#if __has_builtin(__builtin_amdgcn_global_load_async_to_lds_b128)
    if (tid < (M * 3 * 4) / 16) {         // 192 lanes x 16B = 3072 B
        __attribute__((address_space(1))) v4i* g =
            (__attribute__((address_space(1))) v4i*)(void*)(sp + tid * 4);
        __attribute__((address_space(3))) v4i* l =
            (__attribute__((address_space(3))) v4i*)(void*)(s_sp + tid * 4);
        __builtin_amdgcn_global_load_async_to_lds_b128(g, l, 0, 0);
    }
#if __has_builtin(__builtin_amdgcn_s_wait_asynccnt)
    __builtin_amdgcn_s_wait_asynccnt(0);
#else
    asm volatile("s_wait_asynccnt 0x0" ::: "memory");
#endif
#else
    for (int c = tid; c < M * 3; c += 256) s_sp[c] = sp[c];
#endif
    __syncthreads();
    const int i = blockIdx.x * 256 + tid;
    const float x = pc[i * 6 + 0], y = pc[i * 6 + 1], z = pc[i * 6 + 2];
    float best = __builtin_inff(); int bc = 0;
    for (int c = 0; c < M; ++c) {
        const float dx = x - s_sp[c * 3 + 0];
        const float dy = y - s_sp[c * 3 + 1];
        const float dz = z - s_sp[c * 3 + 2];
        const float d  = dx * dx + dy * dy + dz * dz;
        if (d < best) { best = d; bc = c; }
    }
    up[i] = bc;
    atomicAdd(&count[bc], 1);
}

// ---------------------------------------------------------------------------
// K3: level-1 xyz KNN, k=10. One block per query. All 65536 distances are
// computed ONCE into a 256 KB LDS cache (instead of 10 global rescans:
// saves ~3.5 GB of L2 traffic across the 256 blocks); then 10 stable argmin
// passes over LDS, invalidating each selected entry.
// ---------------------------------------------------------------------------
__global__ __launch_bounds__(256)
void knn1_kernel(const float* __restrict__ pc, const float* __restrict__ sp,
                 int* __restrict__ nbr) {
    __shared__ float s_d[N_PTS];          // 256 KB distance cache
    __shared__ float s_v[256];
    __shared__ int   s_i[256];
    const int q = blockIdx.x, tid = threadIdx.x;
    const float qx = sp[q * 3 + 0], qy = sp[q * 3 + 1], qz = sp[q * 3 + 2];
    for (int p = tid; p < N_PTS; p += 256) {
        const float dx = pc[p * 6 + 0] - qx;
        const float dy = pc[p * 6 + 1] - qy;
        const float dz = pc[p * 6 + 2] - qz;
        s_d[p] = dx * dx + dy * dy + dz * dz;
    }
    __syncthreads();
    for (int j = 0; j < KNN1; ++j) {
        float best = __builtin_inff(); int bi = tid;
        for (int p = tid; p < N_PTS; p += 256) {
            const float d = s_d[p];
            if (d < best) { best = d; bi = p; }       // strict < keeps first min
        }
        s_v[tid] = best; s_i[tid] = bi;
        __syncthreads();
        for (int st = 128; st > 0; st >>= 1) {
            if (tid < st) {
                const float v2 = s_v[tid + st]; const int i2 = s_i[tid + st];
                if (v2 < s_v[tid] || (v2 == s_v[tid] && i2 < s_i[tid])) {
                    s_v[tid] = v2; s_i[tid] = i2;
                }
            }
            __syncthreads();
        }
        if (tid == 0) {
            nbr[q * KNN1 + j] = s_i[0];
            s_d[s_i[0]] = __builtin_inff();           // exclude from later passes
        }
        __syncthreads();
    }
}

// ---------------------------------------------------------------------------
// K4: F1[q][:] = mean over 10 neighbors of features; also n2[q] = ||F1[q]||^2
// ---------------------------------------------------------------------------
__global__ __launch_bounds__(256)
void agg1_kernel(const float* __restrict__ feats, const int* __restrict__ nbr,
                 float* __restrict__ F1, float* __restrict__ n2) {
    __shared__ int   s_n[KNN1];
    __shared__ float s_red[256];
    const int q = blockIdx.x, tid = threadIdx.x;
    if (tid < KNN1) s_n[tid] = nbr[q * KNN1 + tid];
    __syncthreads();
    float local = 0.f;
    for (int d = tid; d < FDIM; d += 256) {
        float acc = 0.f;
        for (int j = 0; j < KNN1; ++j) acc += feats[(size_t)s_n[j] * FDIM + d];
        acc *= (1.0f / KNN1);
        F1[q * FDIM + d] = acc;
        local += acc * acc;
    }
    s_red[tid] = local;
    __syncthreads();
    for (int st = 128; st > 0; st >>= 1) {
        if (tid < st) s_red[tid] += s_red[tid + st];
        __syncthreads();
    }
    if (tid == 0) n2[q] = s_red[0];
}

// ---------------------------------------------------------------------------
// K5: tiny index maps. u2[c] = up[idx[c]], tmap[c] = idx[u2[c]].
// ---------------------------------------------------------------------------
__global__ void maps_kernel(const int* __restrict__ idx, const int* __restrict__ up,
                            int* __restrict__ u2, int* __restrict__ tmap) {
    const int c = threadIdx.x;                        // 256 threads
    const int a = up[idx[c]];
    u2[c]   = a;
    tmap[c] = idx[a];
}

// K6: Qg[q][:] = F1[u2[q]][:]  (materialize query rows for the GEMM)
__global__ __launch_bounds__(256)
void gatherq_kernel(const float* __restrict__ F1, const int* __restrict__ u2,
                    float* __restrict__ Qg) {
    const int q = blockIdx.x, tid = threadIdx.x;
    const int src = u2[q];
    for (int d = tid; d < FDIM; d += 256)
        Qg[q * FDIM + d] = F1[src * FDIM + d];
}

// ---------------------------------------------------------------------------
// K7: Gram GEMM  G = Qg (256x768) * F1^T  via fp32 WMMA.
// 16 workgroups (one per 16-row A strip), 512 threads = 16 waves, each wave
// owns one 16x16 C tile (tn = wave id). Wave 0 stages the 48 KB A panel into
// LDS with the Tensor Data Mover (tensor_load_to_lds, D# built per the ISA
// group0/group1 bit layout) + s_wait_tensorcnt; all waves then consume A via
// ds_load_b64 while streaming B rows from L2, K stepped by 4 with
// V_WMMA_F32_16X16X4_F32. A 16x4 f32 layout: lanes 0-15 K=0..1, lanes 16-31
// K=2..3, M = lane&15; B symmetric; C: VGPR v -> M = 8*(lane>>4)+v.
// ---------------------------------------------------------------------------
__global__ __launch_bounds__(512)
void gram_kernel(const float* __restrict__ Qg, const float* __restrict__ F1,
                 float* __restrict__ G) {
    __shared__ float s_A[16 * FDIM];                  // 48 KB A panel
    const int tm   = blockIdx.x;
    const int tid  = threadIdx.x;
    const int wave = tid >> 5;                        // 0..15 = tn
    const int lane = tid & 31;

#if __has_builtin(__builtin_amdgcn_tensor_load_to_lds)
    if (wave == 0) {                                  // one wave issues the DMA
        const unsigned laddr =
            (unsigned)(unsigned long long)(uintptr_t)(void*)s_A;
        const unsigned long long ga =
            (unsigned long long)(uintptr_t)(const void*)(Qg + (size_t)tm * 16 * FDIM);
        // D# group0: count=1 | lds_addr | global_addr[56:0] | type=2
        v4u g0 = { 1u, laddr, (unsigned)ga,
                   (unsigned)((ga >> 32) & 0x1FFFFFFull) | 0x80000000u };
        // D# group1: data_size=4B; tensor_dim0=768, tensor_dim1=16;
        // tile_dim0=768, tile_dim1=16; tensor_dim0_stride=768
        v8i g1 = { (int)(2u << 16),                   // workgroup_mask=0, data_size=2
                   (int)(768u << 16),                 // tensor_dim0[15:0]
                   (int)(16u << 16),                  // tensor_dim0 hi=0, tensor_dim1 lo
                   (int)(768u << 16),                 // tensor_dim1 hi=0, tile_dim0
                   16,                                // tile_dim1, tile_dim2=0
                   768,                               // tensor_dim0_stride[31:0]
                   0, 0 };
        v4i gz = { 0, 0, 0, 0 };
#if defined(__clang_major__) && (__clang_major__ >= 23)
        v8i gz8 = { 0, 0, 0, 0, 0, 0, 0, 0 };
        __builtin_amdgcn_tensor_load_to_lds(g0, g1, gz, gz, gz8, 0);
#else
        __builtin_amdgcn_tensor_load_to_lds(g0, g1, gz, gz, 0);
#endif
#if __has_builtin(__builtin_amdgcn_s_wait_tensorcnt)
        __builtin_amdgcn_s_wait_tensorcnt(0);
#else
        asm volatile("s_wait_tensorcnt 0x0" ::: "memory");
#endif
    }
#else
    for (int d = tid; d < 16 * FDIM; d += 512)
        s_A[d] = Qg[(size_t)tm * 16 * FDIM + d];
#endif
    __syncthreads();

    const int r  = lane & 15;
    const int kh = (lane >> 4) << 1;                  // 0 or 2
    const float* __restrict__ arow = s_A + (size_t)r * FDIM + kh;
    const float* __restrict__ brow = F1 + (size_t)(wave * 16 + r) * FDIM + kh;

    v8f acc = {};
#pragma unroll 8
    for (int k0 = 0; k0 < FDIM; k0 += 4) {
        v2f a = *(const v2f*)(arow + k0);             // ds_load_b64
        v2f b = *(const v2f*)(brow + k0);             // global_load_b64
        acc = __builtin_amdgcn_wmma_f32_16x16x4_f32(
            false, a, false, b, (short)0, acc, false, false);
    }
    const int mrow = (lane >> 4) << 3;
    const int col  = wave * 16 + (lane & 15);
#pragma unroll
    for (int v = 0; v < 8; ++v)
        G[(size_t)(tm * 16 + mrow + v) * M + col] = acc[v];
}

// ---------------------------------------------------------------------------
// K8: level-2 top-90 over duplicated columns. Score s[c] = n2[c] - 2*G[q][c];
// greedy distance-sorted group walk weighted by count[c]; F2 = weighted mean.
// ---------------------------------------------------------------------------
__global__ __launch_bounds__(256)
void select2_kernel(const float* __restrict__ Gm, const float* __restrict__ n2,
                    const int* __restrict__ count, const float* __restrict__ F1,
                    float* __restrict__ F2) {
    __shared__ float s_s[M];
    __shared__ int   s_c[M];
    __shared__ float s_w[M];
    __shared__ int   s_ng;
    const int q = blockIdx.x, tid = threadIdx.x;
    s_s[tid] = n2[tid] - 2.0f * Gm[(size_t)q * M + tid];
    __syncthreads();
    if (tid == 0) {
        int rem = KNN2, ng = 0;
        while (rem > 0) {
            float best = __builtin_inff(); int bc = -1;
            for (int c = 0; c < M; ++c)
                if (s_s[c] < best) { best = s_s[c]; bc = c; }
            if (bc < 0) break;                        // safety (cannot happen)
            int t = count[bc]; if (t > rem) t = rem;
            if (t > 0) { s_c[ng] = bc; s_w[ng] = (float)t; ++ng; rem -= t; }
            s_s[bc] = __builtin_inff();
        }
        s_ng = ng;
    }
    __syncthreads();
    const int ng = s_ng;
    for (int d = tid; d < FDIM; d += 256) {
        float acc = 0.f;
        for (int g = 0; g < ng; ++g) acc += s_w[g] * F1[s_c[g] * FDIM + d];
        F2[q * FDIM + d] = acc * (1.0f / KNN2);
    }
}

// ---------------------------------------------------------------------------
// K9: materialize outputs: pts copy + feats upsample F2[up[i]] (float4).
// ---------------------------------------------------------------------------
__global__ __launch_bounds__(192)
void finalize_feats_kernel(const float* __restrict__ pc, const int* __restrict__ up,
                           const float* __restrict__ F2,
                           float* __restrict__ out_pts, float* __restrict__ out_feats) {
    __shared__ int s_c;
    const int i = blockIdx.x, tid = threadIdx.x;
    if (tid == 0) s_c = up[i];
    __syncthreads();
    const float4* src = (const float4*)(F2 + (size_t)s_c * FDIM);
    float4*       dst = (float4*)(out_feats + (size_t)i * FDIM);
    dst[tid] = src[tid];                              // 192 * 4 = 768
    if (tid < 3) out_pts[i * 3 + tid] = pc[i * 6 + tid];
}

// K10: labels via the collapsed column map tmap[up[i]].
__global__ __launch_bounds__(256)
void finalize_labels_kernel(const float* __restrict__ labels0, const int* __restrict__ up,
                            const int* __restrict__ tmap, float* __restrict__ out_labels) {
    const int g = blockIdx.x * 256 + threadIdx.x;     // 64*65536 threads
    const int l = g >> 16;
    const int i = g & 65535;
    const int col = tmap[up[i]];
    out_labels[(size_t)l * N_PTS + i] = labels0[(size_t)l * N_PTS + col];
}

// ---------------------------------------------------------------------------
extern "C" void kernel_launch(void* const* d_in, const int* in_sizes, int n_in,
                              void* d_out, int out_size, void* d_ws, size_t ws_size,
                              hipStream_t stream) {
    const float* pc       = (const float*)d_in[0];    // (65536, 6)
    const float* features = (const float*)d_in[1];    // (65536, 768)
    const float* labels0  = (const float*)d_in[2];    // (64, 65536)

    float* out        = (float*)d_out;
    float* out_pts    = out;                                   // 65536*3
    float* out_feats  = out + (size_t)N_PTS * 3;               // 65536*768
    float* out_labels = out_feats + (size_t)N_PTS * FDIM;      // 64*65536

    char* w = (char*)d_ws;
    int*   idx   = (int*)w;                w += (size_t)M * 4;
    float* sp    = (float*)w;              w += (size_t)M * 3 * 4;
    int*   nbr1  = (int*)w;                w += (size_t)M * KNN1 * 4;
    int*   up    = (int*)w;                w += (size_t)N_PTS * 4;       // 256 KB
    int*   count = (int*)w;                w += (size_t)M * 4;
    float* F1    = (float*)w;              w += (size_t)M * FDIM * 4;    // 768 KB
    float* n2    = (float*)w;              w += (size_t)M * 4;
    int*   u2    = (int*)w;                w += (size_t)M * 4;
    int*   tmap  = (int*)w;                w += (size_t)M * 4;
    float* Qg    = (float*)w;              w += (size_t)M * FDIM * 4;    // 768 KB
    float* G     = (float*)w;              w += (size_t)M * M * 4;       // 256 KB
    float* F2    = (float*)w;              w += (size_t)M * FDIM * 4;    // 768 KB

    zero_count_kernel<<<1, 256, 0, stream>>>(count);
    fps_kernel<<<1, 1024, 0, stream>>>(pc, idx, sp);
    up_count_kernel<<<N_PTS / 256, 256, 0, stream>>>(pc, sp, up, count);
    knn1_kernel<<<M, 256, 0, stream>>>(pc, sp, nbr1);
    agg1_kernel<<<M, 256, 0, stream>>>(features, nbr1, F1, n2);
    maps_kernel<<<1, 256, 0, stream>>>(idx, up, u2, tmap);
    gatherq_kernel<<<M, 256, 0, stream>>>(F1, u2, Qg);
    gram_kernel<<<M / 16, 512, 0, stream>>>(Qg, F1, G);
    select2_kernel<<<M, 256, 0, stream>>>(G, n2, count, F1, F2);
    finalize_feats_kernel<<<N_PTS, 192, 0, stream>>>(pc, up, F2, out_pts, out_feats);
    finalize_labels_kernel<<<(NLBL * N_PTS) / 256, 256, 0, stream>>>(labels0, up, tmap, out_labels);
}